// LSHAttention_25761213841557
// MI455X (gfx1250) — compile-verified
//
#include <hip/hip_runtime.h>
#include <math.h>

#define B_ 8
#define S_ 8192
#define D_ 64
#define H_ 8
#define NB_ 128           // buckets per hash
#define BS_ 64            // bucket/chunk size
#define CH_ (H_*NB_)      // 1024 chunks per batch
#define SLOTS_ (H_*S_)    // 65536 sorted slots per batch
#define TCH_ 256          // token chunk for stable counting sort
#define NTCH_ (S_/TCH_)   // 32

#define QS_STRIDE 66
#define PS_STRIDE 130

typedef __attribute__((ext_vector_type(2))) float v2f;
typedef __attribute__((ext_vector_type(8))) float v8f;

// ---------------------------------------------------------------------------
// K1: hash_vectors — rotated = qk @ rot, bucket = argmax([r, -r]); fused
//     per-(b,h,tchunk,bucket) histogram for the stable counting sort.
// ---------------------------------------------------------------------------
__global__ void __launch_bounds__(256) k_hash(const float* __restrict__ qk,
                                              const float* __restrict__ rot,
                                              int* __restrict__ buckets,
                                              int* __restrict__ counts) {
  __shared__ float rl[64 * 64];   // rotations slice [f][i] for this hash
  int blk = blockIdx.x;           // (b*H + h)*NTCH + tc
  int tc  = blk % NTCH_;
  int bh  = blk / NTCH_;
  int h   = bh % H_;
  int tid = threadIdx.x;
  int b   = bh / H_;

  for (int i = tid; i < 64 * 64; i += 256) {
    int f = i >> 6, c = i & 63;
    rl[i] = rot[((size_t)f * H_ + h) * 64 + c];
  }
  __syncthreads();

  int t = tc * TCH_ + tid;
  const float* q = qk + ((size_t)b * S_ + t) * D_;
  float qr[64];
#pragma unroll
  for (int f = 0; f < 64; ++f) qr[f] = q[f];

  float bp = -INFINITY, bn = -INFINITY;
  int ip = 0, in_ = 64;
  for (int i = 0; i < 64; ++i) {
    float s = 0.f;
#pragma unroll
    for (int f = 0; f < 64; ++f) s = fmaf(qr[f], rl[f * 64 + i], s);
    if (s > bp)  { bp = s;  ip  = i; }
    if (-s > bn) { bn = -s; in_ = i + 64; }
  }
  int u = (bp >= bn) ? ip : in_;   // first-occurrence argmax semantics
  buckets[(size_t)bh * S_ + t] = u;
  atomicAdd(&counts[((size_t)bh * NTCH_ + tc) * NB_ + u], 1);
}

// ---------------------------------------------------------------------------
// K2: per-(b,h) exclusive scan over (chunk,bucket) histogram -> scatter bases
// ---------------------------------------------------------------------------
__global__ void __launch_bounds__(NB_) k_scan(const int* __restrict__ counts,
                                              int* __restrict__ excl) {
  __shared__ int tot[NB_];
  int bh = blockIdx.x;
  int u  = threadIdx.x;  // 0..127 bucket
  int run = 0;
  for (int c = 0; c < NTCH_; ++c) {
    size_t idx = ((size_t)bh * NTCH_ + c) * NB_ + u;
    int v = counts[idx];
    excl[idx] = run;
    run += v;
  }
  tot[u] = run;
  __syncthreads();
  for (int off = 1; off < NB_; off <<= 1) {   // inclusive Hillis-Steele scan
    int v = (u >= off) ? tot[u - off] : 0;
    __syncthreads();
    tot[u] += v;
    __syncthreads();
  }
  int start = tot[u] - run;                   // exclusive bucket start
  for (int c = 0; c < NTCH_; ++c) {
    size_t idx = ((size_t)bh * NTCH_ + c) * NB_ + u;
    excl[idx] += start;
  }
}

// ---------------------------------------------------------------------------
// K3: stable scatter -> sorted_t (slot -> token), pos_of (token -> slot)
// ---------------------------------------------------------------------------
__global__ void __launch_bounds__(TCH_) k_scatter(const int* __restrict__ buckets,
                                                  const int* __restrict__ excl,
                                                  int* __restrict__ sorted_t,
                                                  int* __restrict__ pos_of) {
  __shared__ int ub[TCH_];
  int blk = blockIdx.x;  // bh*NTCH + tc
  int tc  = blk % NTCH_;
  int bh  = blk / NTCH_;
  int h   = bh % H_;
  int b   = bh / H_;
  int tid = threadIdx.x;
  int t = tc * TCH_ + tid;
  int u = buckets[(size_t)bh * S_ + t];
  ub[tid] = u;
  __syncthreads();
  int rank = 0;
  for (int j = 0; j < tid; ++j) rank += (ub[j] == u);  // stable within chunk
  int pos = excl[((size_t)bh * NTCH_ + tc) * NB_ + u] + rank;
  sorted_t[(size_t)b * SLOTS_ + h * S_ + pos] = t;
  pos_of[(size_t)bh * S_ + t] = h * S_ + pos;
}

// ---------------------------------------------------------------------------
// K4: bucketed attention. One block (4 waves) per (b, chunk). fp32 WMMA.
// ---------------------------------------------------------------------------
__global__ void __launch_bounds__(128) k_attn(const float* __restrict__ qk,
                                              const float* __restrict__ vv,
                                              const int* __restrict__ sorted_t,
                                              float* __restrict__ so,
                                              float* __restrict__ slog) {
  extern __shared__ float smem[];
  int*   tq = (int*)smem;                    // 64 query token ids
  int*   tk = (int*)(smem + 64);             // 128 key token ids
  float* qs = smem + 192;                    // [64][QS_STRIDE]  raw queries
  float* ks = qs + 64 * QS_STRIDE;           // [128][QS_STRIDE] normalized keys
  float* vs = ks + 128 * QS_STRIDE;          // [128][QS_STRIDE] values
  float* ps = vs + 128 * QS_STRIDE;          // [64][PS_STRIDE]  probs

  int c  = blockIdx.x % CH_;
  int b  = blockIdx.x / CH_;
  int cp = (c + CH_ - 1) % CH_;              // look-one-back chunk (wraps)
  int tid  = threadIdx.x;                    // 128 threads
  int lane = tid & 31;
  int wave = tid >> 5;                       // 0..3 -> M tile
  int laneN = lane & 15;
  int half  = lane >> 4;

  // ---- stage LDS: 64 q rows, 128 normalized k rows, 128 v rows ----
  {
    int j = tid;
    int tok = (j < 64) ? sorted_t[(size_t)b * SLOTS_ + (size_t)c * BS_ + j]
                       : sorted_t[(size_t)b * SLOTS_ + (size_t)cp * BS_ + (j - 64)];
    tk[j] = tok;
    if (j < 64) tq[j] = tok;

    const float4* qrow = (const float4*)(qk + ((size_t)b * S_ + tok) * D_);
    float r[64];
    float ss = 0.f;
#pragma unroll
    for (int q4 = 0; q4 < 16; ++q4) {
      float4 x = qrow[q4];
      r[q4 * 4 + 0] = x.x; r[q4 * 4 + 1] = x.y;
      r[q4 * 4 + 2] = x.z; r[q4 * 4 + 3] = x.w;
      ss += x.x * x.x + x.y * x.y + x.z * x.z + x.w * x.w;
    }
    float sc = 1.0f / fmaxf(sqrtf(ss), 1e-12f);
    if (j < 64) {
#pragma unroll
      for (int d = 0; d < 64; ++d) qs[j * QS_STRIDE + d] = r[d];
    }
#pragma unroll
    for (int d = 0; d < 64; ++d) ks[j * QS_STRIDE + d] = r[d] * sc;

    const float4* vrow = (const float4*)(vv + ((size_t)b * S_ + tok) * D_);
#pragma unroll
    for (int q4 = 0; q4 < 16; ++q4) {
      float4 x = vrow[q4];
      vs[j * QS_STRIDE + q4 * 4 + 0] = x.x;
      vs[j * QS_STRIDE + q4 * 4 + 1] = x.y;
      vs[j * QS_STRIDE + q4 * 4 + 2] = x.z;
      vs[j * QS_STRIDE + q4 * 4 + 3] = x.w;
    }
  }
  __syncthreads();

  // ---- GEMM 1: dots[64x128] = bq @ bk^T, fp32 WMMA 16x16x4 ----
  const v8f zero8 = {0.f, 0.f, 0.f, 0.f, 0.f, 0.f, 0.f, 0.f};
  v8f acc[8];
#pragma unroll
  for (int n = 0; n < 8; ++n) acc[n] = zero8;

  int mrow = wave * 16 + laneN;              // A-fragment row for this lane
  for (int kk = 0; kk < 16; ++kk) {
    int kb = kk * 4 + half * 2;              // K = kb, kb+1 for this lane
    v2f a;
    a.x = qs[mrow * QS_STRIDE + kb];
    a.y = qs[mrow * QS_STRIDE + kb + 1];
#pragma unroll
    for (int n = 0; n < 8; ++n) {
      v2f bm;                                // B element (K, N) = bk[N][K]
      bm.x = ks[(n * 16 + laneN) * QS_STRIDE + kb];
      bm.y = ks[(n * 16 + laneN) * QS_STRIDE + kb + 1];
      acc[n] = __builtin_amdgcn_wmma_f32_16x16x4_f32(
          false, a, false, bm, (short)0, acc[n], false, false);
    }
  }

  // ---- scale, self-mask, row softmax (C layout: M = r + 8*half, N = lane%16) ----
  int tqv[8], tkv[8];
#pragma unroll
  for (int r = 0; r < 8; ++r) tqv[r] = tq[wave * 16 + r + 8 * half];
#pragma unroll
  for (int n = 0; n < 8; ++n) tkv[n] = tk[n * 16 + laneN];
#pragma unroll
  for (int n = 0; n < 8; ++n) {
#pragma unroll
    for (int r = 0; r < 8; ++r) {
      float x = acc[n][r] * 0.125f;          // D^-0.5 = 1/8
      if (tqv[r] == tkv[n]) x = -5.0e4f;     // TOKEN_SELF_ATTN_VALUE
      acc[n][r] = x;
    }
  }
#pragma unroll
  for (int r = 0; r < 8; ++r) {
    float m = acc[0][r];
#pragma unroll
    for (int n = 1; n < 8; ++n) m = fmaxf(m, acc[n][r]);
    for (int msk = 1; msk < 16; msk <<= 1) m = fmaxf(m, __shfl_xor(m, msk, 32));
    float s = 0.f;
#pragma unroll
    for (int n = 0; n < 8; ++n) s += expf(acc[n][r] - m);
    for (int msk = 1; msk < 16; msk <<= 1) s += __shfl_xor(s, msk, 32);
    float l = m + logf(s);
    int row = wave * 16 + r + 8 * half;
#pragma unroll
    for (int n = 0; n < 8; ++n)
      ps[row * PS_STRIDE + n * 16 + laneN] = expf(acc[n][r] - l);
    if (laneN == 0)
      slog[(size_t)b * SLOTS_ + (size_t)c * BS_ + row] = l;
  }
  __syncthreads();

  // ---- GEMM 2: out[64x64] = probs[64x128] @ bv[128x64] ----
  v8f acc2[4];
#pragma unroll
  for (int n = 0; n < 4; ++n) acc2[n] = zero8;
  for (int kk = 0; kk < 32; ++kk) {
    int kb = kk * 4 + half * 2;
    v2f a;
    a.x = ps[mrow * PS_STRIDE + kb];
    a.y = ps[mrow * PS_STRIDE + kb + 1];
#pragma unroll
    for (int n = 0; n < 4; ++n) {
      v2f bm;                                // B element (K, N) = bv[K][N]
      bm.x = vs[kb * QS_STRIDE + n * 16 + laneN];
      bm.y = vs[(kb + 1) * QS_STRIDE + n * 16 + laneN];
      acc2[n] = __builtin_amdgcn_wmma_f32_16x16x4_f32(
          false, a, false, bm, (short)0, acc2[n], false, false);
    }
  }
#pragma unroll
  for (int n = 0; n < 4; ++n) {
#pragma unroll
    for (int r = 0; r < 8; ++r) {
      int row = wave * 16 + r + 8 * half;
      so[((size_t)b * SLOTS_ + (size_t)c * BS_ + row) * D_ + n * 16 + laneN] =
          acc2[n][r];
    }
  }
}

// ---------------------------------------------------------------------------
// K5: unsort + combine rounds weighted by per-round logsumexp
// ---------------------------------------------------------------------------
__global__ void __launch_bounds__(64) k_combine(const float* __restrict__ so,
                                                const float* __restrict__ slog,
                                                const int* __restrict__ pos_of,
                                                float* __restrict__ out) {
  __shared__ float lg[8];
  __shared__ int ph[8];
  int t = blockIdx.x % S_;
  int b = blockIdx.x / S_;
  int d = threadIdx.x;
  if (d < 8) {
    int p = pos_of[((size_t)(b * H_ + d)) * S_ + t];
    ph[d] = p;
    lg[d] = slog[(size_t)b * SLOTS_ + p];
  }
  __syncthreads();
  float m = lg[0];
#pragma unroll
  for (int h = 1; h < 8; ++h) m = fmaxf(m, lg[h]);
  float w[8];
  float s = 0.f;
#pragma unroll
  for (int h = 0; h < 8; ++h) { w[h] = expf(lg[h] - m); s += w[h]; }
  float inv = 1.0f / s;
  float o = 0.f;
#pragma unroll
  for (int h = 0; h < 8; ++h)
    o += w[h] * inv * so[((size_t)b * SLOTS_ + ph[h]) * D_ + d];
  out[((size_t)b * S_ + t) * D_ + d] = o;
}

// ---------------------------------------------------------------------------
extern "C" void kernel_launch(void* const* d_in, const int* in_sizes, int n_in,
                              void* d_out, int out_size, void* d_ws, size_t ws_size,
                              hipStream_t stream) {
  const float* qk  = (const float*)d_in[0];
  const float* v   = (const float*)d_in[1];
  const float* rot = (const float*)d_in[2];

  char* ws = (char*)d_ws;
  size_t off = 0;
  auto alloc = [&](size_t bytes) -> char* {
    char* p = ws + off;
    off = (off + bytes + 255) & ~(size_t)255;
    return p;
  };
  int*   buckets = (int*)alloc((size_t)B_ * H_ * S_ * 4);           // 2 MB
  int*   counts  = (int*)alloc((size_t)B_ * H_ * NTCH_ * NB_ * 4);  // 1 MB
  int*   excl    = (int*)alloc((size_t)B_ * H_ * NTCH_ * NB_ * 4);  // 1 MB
  int*   sorted  = (int*)alloc((size_t)B_ * SLOTS_ * 4);            // 2 MB
  int*   pos_of  = (int*)alloc((size_t)B_ * H_ * S_ * 4);           // 2 MB
  float* slog    = (float*)alloc((size_t)B_ * SLOTS_ * 4);          // 2 MB
  float* so      = (float*)alloc((size_t)B_ * SLOTS_ * D_ * 4);     // 128 MB

  hipMemsetAsync(counts, 0, (size_t)B_ * H_ * NTCH_ * NB_ * 4, stream);

  k_hash<<<B_ * H_ * NTCH_, 256, 0, stream>>>(qk, rot, buckets, counts);
  k_scan<<<B_ * H_, NB_, 0, stream>>>(counts, excl);
  k_scatter<<<B_ * H_ * NTCH_, TCH_, 0, stream>>>(buckets, excl, sorted, pos_of);

  size_t smem =
      (size_t)(192 + 64 * QS_STRIDE + 2 * 128 * QS_STRIDE + 64 * PS_STRIDE) * 4;
  hipFuncSetAttribute((const void*)k_attn,
                      hipFuncAttributeMaxDynamicSharedMemorySize, (int)smem);
  k_attn<<<B_ * CH_, 128, smem, stream>>>(qk, v, sorted, so, slog);

  k_combine<<<B_ * S_, 64, 0, stream>>>(so, slog, pos_of, (float*)d_out);
}